// GCNv2_35115652612099
// MI455X (gfx1250) — compile-verified
//
#include <hip/hip_runtime.h>

typedef __attribute__((ext_vector_type(2))) float v2f;
typedef __attribute__((ext_vector_type(8))) float v8f;

// ---------------- degree / normalization ----------------

__global__ void deg_init_kernel(float* deg, int n) {
  int i = blockIdx.x * blockDim.x + threadIdx.x;
  if (i < n) deg[i] = 1.0f;  // self-loop contribution
}

__global__ void deg_edges_kernel(const int* __restrict__ dst, float* deg, int e) {
  int i = blockIdx.x * blockDim.x + threadIdx.x;
  if (i < e) atomicAdd(&deg[dst[i]], 1.0f);
}

__global__ void deg_to_dinv_kernel(float* deg, int n) {
  int i = blockIdx.x * blockDim.x + threadIdx.x;
  if (i < n) deg[i] = rsqrtf(deg[i]);  // deg >= 1 always (self-loop)
}

// ---------------- aggregation: agg = D^-1/2 (A+I) D^-1/2 h ----------------

// self-loop term initializes agg (no atomics needed; runs before edge kernel)
__global__ void agg_init_kernel(float* __restrict__ agg, const float* __restrict__ dinv,
                                const float* __restrict__ h, int n) {
  int i = blockIdx.x * blockDim.x + threadIdx.x;  // over n*64
  if (i < n * 64) {
    int node = i >> 6;
    float di = dinv[node];
    agg[i] = di * di * h[i];
  }
}

// one wave32 per edge: lane l handles channels 2l, 2l+1 (float2 = 8B aligned,
// h rows are 256B aligned). Gathers + atomics stay resident in the 192MB L2.
__global__ void agg_edges_kernel(const int* __restrict__ src, const int* __restrict__ dst,
                                 const float* __restrict__ dinv, const float* __restrict__ h,
                                 float* __restrict__ agg, int e) {
  int gid = blockIdx.x * blockDim.x + threadIdx.x;
  int edge = gid >> 5;
  int lane = gid & 31;
  if (edge >= e) return;
  int s = src[edge];
  int d = dst[edge];
  float nrm = dinv[s] * dinv[d];
  float2 v = ((const float2*)(h + (size_t)s * 64))[lane];
  float* ap = agg + (size_t)d * 64 + lane * 2;
  atomicAdd(ap,     nrm * v.x);
  atomicAdd(ap + 1, nrm * v.y);
}

// ---------------- fused (ca*A + cb*B) @ W [+bias] [+ELU], Ncols = 64 ----------------
// One wave computes a 16-row x 64-col tile: 4 accumulator tiles of 16x16 f32,
// K stepped by 4 with V_WMMA_F32_16X16X4_F32 (full fp32 precision; GEMMs here
// are bandwidth-bound so the f32 matrix path is the right precision choice).
// K / flags are compile-time so the hot loop carries no branches or flag tests.
template <int KT, bool HAS_B, bool HAS_BIAS, bool DO_ELU>
__global__ void gemm_n64_wmma_kernel(const float* __restrict__ A, const float* __restrict__ B,
                                     float ca, float cb,
                                     const float* __restrict__ W, const float* __restrict__ bias,
                                     float* __restrict__ out, int n_rows) {
  int wave = (int)((blockIdx.x * blockDim.x + threadIdx.x) >> 5);
  int row0 = wave * 16;
  if (row0 >= n_rows) return;   // wave-uniform exit: EXEC stays all-ones for WMMA
  int lane = threadIdx.x & 31;
  int r    = lane & 15;         // A: row M; B/C/D: col within 16-tile
  int half = lane >> 4;         // selects K pair (A/B) and M+8 (C/D)

  const float* pa = A + (size_t)(row0 + r) * KT + 2 * half;
  const float* pb = HAS_B ? (B + (size_t)(row0 + r) * KT + 2 * half) : pa;

  v8f acc[4] = {};

#pragma unroll(KT == 64 ? 16 : 4)
  for (int k = 0; k < KT; k += 4) {
    // A operand: lane holds (row r, K = k+2*half .. +1)
    float2 av = *(const float2*)(pa + k);
    float s0 = ca * av.x;
    float s1 = ca * av.y;
    if (HAS_B) {
      float2 bv = *(const float2*)(pb + k);
      s0 += cb * bv.x;
      s1 += cb * bv.y;
    }
    v2f amat;
    amat.x = s0;
    amat.y = s1;

    int col = k + 2 * half;                     // K row of W for this lane
    const float* w0 = W + (size_t)col * 64 + r; // W[col][tile*16 + r]
    const float* w1 = w0 + 64;                  // W[col+1][...]
#pragma unroll
    for (int t = 0; t < 4; ++t) {
      v2f bmat;
      bmat.x = w0[t * 16];
      bmat.y = w1[t * 16];
      acc[t] = __builtin_amdgcn_wmma_f32_16x16x4_f32(
          /*neg_a=*/false, amat, /*neg_b=*/false, bmat,
          /*c_mod=*/(short)0, acc[t], /*reuse_a=*/false, /*reuse_b=*/false);
    }
  }

  // epilogue: C/D layout -> element acc[t][rr] lives at (row0+rr+8*half, t*16+r)
#pragma unroll
  for (int t = 0; t < 4; ++t) {
    int c = t * 16 + r;
    float bv = HAS_BIAS ? bias[c] : 0.0f;
#pragma unroll
    for (int rr = 0; rr < 8; ++rr) {
      int m = row0 + rr + 8 * half;
      float val = acc[t][rr] + bv;
      if (DO_ELU) val = (val > 0.0f) ? val : (expf(val) - 1.0f);
      out[(size_t)m * 64 + c] = val;
    }
  }
}

// ---------------- launch ----------------

extern "C" void kernel_launch(void* const* d_in, const int* in_sizes, int n_in,
                              void* d_out, int out_size, void* d_ws, size_t ws_size,
                              hipStream_t stream) {
  const float* x     = (const float*)d_in[0];  // [N, 256]
  const int*   ei    = (const int*)  d_in[1];  // [2, E]
  const float* W_fc  = (const float*)d_in[2];  // [256, 64]
  const float* b_fc  = (const float*)d_in[3];  // [64]
  const float* W1    = (const float*)d_in[4];  // [4, 64, 64]
  const float* W_out = (const float*)d_in[5];  // [64, 64]
  const float* b_out = (const float*)d_in[6];  // [64]

  const int IN = 256, H = 64, L = 4;
  const int N = in_sizes[0] / IN;
  const int E = in_sizes[1] / 2;
  const int* src = ei;
  const int* dst = ei + E;

  float* ws   = (float*)d_ws;
  float* dinv = ws;                     // [N]
  float* x0   = dinv + N;               // [N, 64]
  float* h    = x0 + (size_t)N * H;     // [N, 64]
  float* agg  = h  + (size_t)N * H;     // [N, 64]

  const int tb = 256;

  // symmetric normalization: dinv = rsqrt(deg(A+I))
  deg_init_kernel<<<(N + tb - 1) / tb, tb, 0, stream>>>(dinv, N);
  deg_edges_kernel<<<(E + tb - 1) / tb, tb, 0, stream>>>(dst, dinv, E);
  deg_to_dinv_kernel<<<(N + tb - 1) / tb, tb, 0, stream>>>(dinv, N);

  // h = x0 = x @ W_fc + b_fc   (K = 256, +bias, no ELU)
  const int waves = (N + 15) / 16;
  const int gemm_blocks = (waves * 32 + tb - 1) / tb;
  gemm_n64_wmma_kernel<256, false, true, false><<<gemm_blocks, tb, 0, stream>>>(
      x, nullptr, 1.0f, 0.0f, W_fc, b_fc, x0, N);

  for (int i = 0; i < L; ++i) {
    const float* hp = (i == 0) ? x0 : h;
    // agg = norm-weighted aggregation of hp
    agg_init_kernel<<<((N * 64) + tb - 1) / tb, tb, 0, stream>>>(agg, dinv, hp, N);
    agg_edges_kernel<<<(int)(((size_t)E * 32 + tb - 1) / tb), tb, 0, stream>>>(
        src, dst, dinv, hp, agg, E);
    // h = elu(((1-a)*agg + a*x0) @ W1[i])   (K = 64, support combine fused)
    float alpha = (float)i / (float)L;
    gemm_n64_wmma_kernel<64, true, false, true><<<gemm_blocks, tb, 0, stream>>>(
        agg, x0, 1.0f - alpha, alpha, W1 + (size_t)i * H * H, nullptr, h, N);
  }

  // out = h @ W_out + b_out   (K = 64, +bias, no ELU)
  gemm_n64_wmma_kernel<64, false, true, false><<<gemm_blocks, tb, 0, stream>>>(
      h, nullptr, 1.0f, 0.0f, W_out, b_out, (float*)d_out, N);
}